// HamiltonianSystem_62929860821510
// MI455X (gfx1250) — compile-verified
//
#include <hip/hip_runtime.h>
#include <hip/hip_bf16.h>

// ---------------------------------------------------------------------------
// HamiltonianSystem: dq_dt = momentum @ inv(M)^T   (fp32 WMMA GEMM)
//                    dp_dt = -position             (analytic central FD of 0.5*sum(q^2))
// B=128, D=64, fp32 everywhere. Problem is L2-resident and latency-bound;
// the GEMM uses the CDNA5 fp32 matrix pipe (V_WMMA_F32_16X16X4_F32) to keep
// full precision while exercising the WMMA path.
// ---------------------------------------------------------------------------

typedef __attribute__((ext_vector_type(2))) float v2f;
typedef __attribute__((ext_vector_type(8))) float v8f;

#define BDIM 128
#define DDIM 64

// ---------------------------------------------------------------------------
// Kernel 1: Gauss-Jordan inversion of the 64x64 mass matrix in LDS.
// One workgroup, 64 threads (2 waves). Augmented matrix [64][128] = 32 KB LDS.
// Thread r owns row r for elimination; all 64 threads cooperate on row-scaling.
// (Mass matrices are SPD; no pivoting needed for the well-conditioned case.)
// ---------------------------------------------------------------------------
__global__ void __launch_bounds__(64)
hs_invert64_kernel(const float* __restrict__ m, float* __restrict__ minv) {
    __shared__ float aug[DDIM][2 * DDIM];
    const int r = threadIdx.x;  // 0..63

    // Init augmented matrix [M | I]
    for (int j = 0; j < DDIM; ++j) {
        aug[r][j]        = m[r * DDIM + j];
        aug[r][DDIM + j] = (r == j) ? 1.0f : 0.0f;
    }
    __syncthreads();

    for (int k = 0; k < DDIM; ++k) {
        // Every thread snapshots the pivot before anyone scales row k.
        const float pinv = 1.0f / aug[k][k];
        __syncthreads();
        // Cooperative scale of row k: thread t handles columns t and t+64.
        aug[k][r]        *= pinv;
        aug[k][DDIM + r] *= pinv;
        __syncthreads();
        // Eliminate column k from every other row (thread r owns row r).
        if (r != k) {
            const float f = aug[r][k];
            #pragma unroll 4
            for (int j = 0; j < 2 * DDIM; ++j) {
                aug[r][j] -= f * aug[k][j];
            }
        }
        __syncthreads();
    }

    // Right half is inv(M); write row-major.
    for (int j = 0; j < DDIM; ++j) {
        minv[r * DDIM + j] = aug[r][DDIM + j];
    }
}

// ---------------------------------------------------------------------------
// Kernel 2: dq = momentum[128,64] @ Minv^T[64,64] using V_WMMA_F32_16X16X4_F32.
// One wave32 per 16x16 output tile; grid = (128/16, 64/16) = (8,4).
// K-loop: 64/4 = 16 chained WMMAs accumulating in the 8-VGPR C/D fragment.
//
// Fragment layouts (ISA 7.12.2):
//  A (16x4 f32, 2 VGPRs): lanes 0-15 -> {K=0,K=1}, lanes 16-31 -> {K=2,K=3},
//                          M = lane & 15.
//  B (4x16 f32, 2 VGPRs): N = lane & 15; lanes 0-15 -> {K=0,K=1},
//                          lanes 16-31 -> {K=2,K=3}.
//  C/D (16x16 f32, 8 VGPRs): VGPR v, lane L -> row = v + (L>=16 ? 8 : 0),
//                             col = L & 15.
// B = Minv^T  =>  B[k][n] = Minv[n][k]: contiguous along k in row-major Minv,
// so both A and B fragments are 8-byte-aligned float2 (global_load_b64) loads.
// ---------------------------------------------------------------------------
__global__ void __launch_bounds__(32)
hs_gemm_wmma_kernel(const float* __restrict__ mom,
                    const float* __restrict__ minv,
                    float* __restrict__ dq) {
    const int lane = threadIdx.x & 31;
    const int half = lane >> 4;       // 0: lanes 0-15, 1: lanes 16-31
    const int l16  = lane & 15;
    const int mt   = blockIdx.x;      // 0..7  (rows of momentum)
    const int nt   = blockIdx.y;      // 0..3  (cols of output)

    const int arow = mt * 16 + l16;   // momentum row this lane feeds into A
    const int bcol = nt * 16 + l16;   // output col this lane feeds into B

    v8f acc = {};
    #pragma unroll
    for (int k0 = 0; k0 < DDIM; k0 += 4) {
        const int ka = k0 + half * 2;  // even -> 8B aligned
        v2f a = *(const v2f*)(mom  + arow * DDIM + ka);
        v2f b = *(const v2f*)(minv + bcol * DDIM + ka);  // Minv[n][k] == (Minv^T)[k][n]
        // 8 args: (neg_a, A, neg_b, B, c_mod, C, reuse_a, reuse_b)
        acc = __builtin_amdgcn_wmma_f32_16x16x4_f32(
            false, a, false, b, (short)0, acc, false, false);
    }

    #pragma unroll
    for (int v = 0; v < 8; ++v) {
        const int orow = mt * 16 + v + half * 8;
        const int ocol = nt * 16 + l16;
        dq[orow * DDIM + ocol] = acc[v];
    }
}

// ---------------------------------------------------------------------------
// Kernel 3: dp_dt = -position. The central difference of U=0.5*sum(q^2) with
// step eps is exactly ((q+eps)^2-(q-eps)^2)/(4*eps) = q, so dp = -q.
// ---------------------------------------------------------------------------
__global__ void __launch_bounds__(256)
hs_negate_kernel(const float* __restrict__ pos, float* __restrict__ dp, int n) {
    const int i = blockIdx.x * blockDim.x + threadIdx.x;
    if (i < n) dp[i] = -pos[i];
}

extern "C" void kernel_launch(void* const* d_in, const int* in_sizes, int n_in,
                              void* d_out, int out_size, void* d_ws, size_t ws_size,
                              hipStream_t stream) {
    const float* position = (const float*)d_in[0];   // [128,64]
    const float* momentum = (const float*)d_in[1];   // [128,64]
    const float* mass     = (const float*)d_in[2];   // [64,64]

    float* dq   = (float*)d_out;             // first 8192 floats
    float* dp   = dq + BDIM * DDIM;          // second 8192 floats
    float* minv = (float*)d_ws;              // 64*64*4 = 16 KB scratch

    hs_invert64_kernel<<<1, 64, 0, stream>>>(mass, minv);

    dim3 ggrid(BDIM / 16, DDIM / 16);        // (8,4) tiles, one wave32 each
    hs_gemm_wmma_kernel<<<ggrid, 32, 0, stream>>>(momentum, minv, dq);

    const int n = BDIM * DDIM;
    hs_negate_kernel<<<(n + 255) / 256, 256, 0, stream>>>(position, dp, n);
}